// DeformableConv2dBnReLU_50895362458096
// MI455X (gfx1250) — compile-verified
//
#include <hip/hip_runtime.h>
#include <hip/hip_bf16.h>

typedef float v2f __attribute__((ext_vector_type(2)));
typedef float v8f __attribute__((ext_vector_type(8)));
typedef int v4i __attribute__((vector_size(16)));

#define HW 25600      // 160*160
#define CIN 64
#define COUT 64
#define IMH 160
#define IMW 160
#define NB 4
#define MTOT (NB * HW)              // 102400
#define OUT_ELEMS (NB * COUT * HW)  // 6553600
#define OFF_ELEMS (NB * 18 * HW)    // 1843200
#define ASTRIDE 68                  // padded LDS row stride for A tile

#ifndef USE_ASYNC_LDS
#if defined(__has_builtin)
#if __has_builtin(__builtin_amdgcn_global_load_async_to_lds_b128)
#define USE_ASYNC_LDS 1
#endif
#endif
#endif
#ifndef USE_ASYNC_LDS
#define USE_ASYNC_LDS 0
#endif

// ---------------- Kernel 0: x NCHW -> NHWC ----------------
__global__ void k_transpose_nhwc(const float* __restrict__ x, float* __restrict__ xh) {
    int o = blockIdx.x * 256 + threadIdx.x;
    if (o >= OUT_ELEMS) return;
    int c = o & 63;
    int rest = o >> 6;           // b*HW + h*W + w
    int w = rest % IMW;
    int t = rest / IMW;
    int h = t % IMH;
    int b = t / IMH;
    xh[o] = x[((b * CIN + c) * IMH + h) * IMW + w];
}

// ---------------- Kernel 1: conv_w -> pair-interleaved Wt ----------------
// Wt[t*4096 + ((cin>>1)*64 + oc)*2 + (cin&1)] = conv_w[oc][cin][tap t]
// so a WMMA B-fragment (rows k,k+1 at column oc) is one aligned float2 in LDS.
__global__ void k_transpose_w(const float* __restrict__ cw, float* __restrict__ wt) {
    int i = blockIdx.x * 256 + threadIdx.x;
    if (i >= 576 * 64) return;
    int t = i / 4096;
    int r = i % 4096;
    int e = r & 1;
    int j = r >> 1;              // 0..2047
    int oc = j & 63;
    int q = j >> 6;              // cin pair 0..31
    int cin = q * 2 + e;
    wt[i] = cw[(oc * CIN + cin) * 9 + t];
}

// ---------------- Kernel 2: offset conv + clamp ----------------
__global__ void k_offset_conv(const float* __restrict__ d, const float* __restrict__ ow,
                              float* __restrict__ off) {
    __shared__ float sw[576];
    int idx = blockIdx.x * 256 + threadIdx.x;
    int w = idx % IMW;
    int h = (idx / IMW) % IMH;
    int ch = (idx / HW) % 18;    // uniform within block (HW % 256 == 0)
    int b = idx / (18 * HW);
    for (int i = threadIdx.x; i < 576; i += 256) sw[i] = ow[ch * 576 + i];
    __syncthreads();
    float acc = 0.0f;
    const float* db = d + b * CIN * HW;
    #pragma unroll
    for (int kh = 0; kh < 3; ++kh) {
        int y = h + kh - 1;
        if (y < 0 || y >= IMH) continue;
        #pragma unroll
        for (int kw = 0; kw < 3; ++kw) {
            int x = w + kw - 1;
            if (x < 0 || x >= IMW) continue;
            const float* dp = db + y * IMW + x;
            const float* wp = sw + (kh * 3 + kw);
            #pragma unroll 8
            for (int c = 0; c < CIN; ++c)
                acc = fmaf(dp[c * HW], wp[c * 9], acc);
        }
    }
    acc = fminf(fmaxf(acc, -1.0f), 1.0f);
    off[idx] = acc;
}

// ---------------- Kernel 3: fused bilinear sampling + WMMA implicit GEMM ----------------
// Each WG: 128 pixels x 64 out-channels. 8 waves, each wave a 16x64 sub-tile.
__global__ void k_deform_gemm(const float* __restrict__ xh, const float* __restrict__ wt,
                              const float* __restrict__ off, float* __restrict__ out) {
    extern __shared__ float smem[];
    float* sW = smem;                 // [32 pairs][64 oc][2] interleaved weight chunk
    float* sA = smem + 4096;          // [128][ASTRIDE] sampled A tile

    const int tid = threadIdx.x;
    const int lane = tid & 31;
    const int wid = tid >> 5;
    const int ln = lane & 15;
    const int hi = lane >> 4;

    // sampling role
    const int p = tid >> 1;            // pixel 0..127
    const int cbase = (tid & 1) * 32;  // channel half
    const int mg = blockIdx.x * 128 + p;
    const int sb = mg / HW;
    const int srr = mg % HW;
    const int sh = srr / IMW;
    const int sw_ = srr % IMW;
    const float* offB = off + sb * 18 * HW + srr;
    const float* xb = xh + sb * (HW * CIN);

    v8f acc[4] = {};

    for (int t = 0; t < 9; ++t) {
        __syncthreads();
        // stage weight chunk for tap t (16 KB flat copy; interleave baked into Wt)
#if USE_ASYNC_LDS
        {
            __attribute__((address_space(1))) char* g =
                (__attribute__((address_space(1))) char*)(wt + t * 4096);
            __attribute__((address_space(3))) char* l =
                (__attribute__((address_space(3))) char*)sW;
            int boff = tid * 16;
            #pragma unroll
            for (int j = 0; j < 4; ++j)
                __builtin_amdgcn_global_load_async_to_lds_b128(
                    (__attribute__((address_space(1))) v4i*)(g + boff + j * 4096),
                    (__attribute__((address_space(3))) v4i*)(l + boff + j * 4096),
                    0, 0);
        }
#else
        {
            const float4* src = reinterpret_cast<const float4*>(wt + t * 4096);
            float4* dst = reinterpret_cast<float4*>(sW);
            #pragma unroll
            for (int j = 0; j < 4; ++j) dst[tid * 4 + j] = src[tid * 4 + j];
        }
#endif
        // bilinear sample 32 channels for this thread's pixel into sA
        {
            float dy = offB[(2 * t) * HW];
            float dx = offB[(2 * t + 1) * HW];
            int kh = t / 3, kw = t % 3;
            float py = (float)(sh - 1 + kh) + dy;
            float px = (float)(sw_ - 1 + kw) + dx;
            float y0f = floorf(py), x0f = floorf(px);
            int y0 = (int)y0f, x0 = (int)x0f;
            float wy = py - y0f, wx = px - x0f;
            float w00 = (1.0f - wy) * (1.0f - wx);
            float w01 = (1.0f - wy) * wx;
            float w10 = wy * (1.0f - wx);
            float w11 = wy * wx;
            int y1 = y0 + 1, x1 = x0 + 1;
            bool vy0 = (y0 >= 0) & (y0 < IMH), vy1 = (y1 >= 0) & (y1 < IMH);
            bool vx0 = (x0 >= 0) & (x0 < IMW), vx1 = (x1 >= 0) & (x1 < IMW);
            if (!(vy0 & vx0)) w00 = 0.0f;
            if (!(vy0 & vx1)) w01 = 0.0f;
            if (!(vy1 & vx0)) w10 = 0.0f;
            if (!(vy1 & vx1)) w11 = 0.0f;
            int y0c = min(max(y0, 0), IMH - 1), y1c = min(max(y1, 0), IMH - 1);
            int x0c = min(max(x0, 0), IMW - 1), x1c = min(max(x1, 0), IMW - 1);
            const float4* q00 = reinterpret_cast<const float4*>(xb + (y0c * IMW + x0c) * CIN + cbase);
            const float4* q01 = reinterpret_cast<const float4*>(xb + (y0c * IMW + x1c) * CIN + cbase);
            const float4* q10 = reinterpret_cast<const float4*>(xb + (y1c * IMW + x0c) * CIN + cbase);
            const float4* q11 = reinterpret_cast<const float4*>(xb + (y1c * IMW + x1c) * CIN + cbase);
            float4* dst = reinterpret_cast<float4*>(&sA[p * ASTRIDE + cbase]);
            #pragma unroll
            for (int cc = 0; cc < 8; ++cc) {
                float4 a = q00[cc], b4 = q01[cc], c4 = q10[cc], d4 = q11[cc];
                float4 r;
                r.x = w00 * a.x + w01 * b4.x + w10 * c4.x + w11 * d4.x;
                r.y = w00 * a.y + w01 * b4.y + w10 * c4.y + w11 * d4.y;
                r.z = w00 * a.z + w01 * b4.z + w10 * c4.z + w11 * d4.z;
                r.w = w00 * a.w + w01 * b4.w + w10 * c4.w + w11 * d4.w;
                dst[cc] = r;
            }
        }
#if USE_ASYNC_LDS
        asm volatile("s_wait_asynccnt 0" ::: "memory");
#endif
        __syncthreads();

        // WMMA over this tap's K=64 slice: 16 steps of K=4
        const float* aRow = &sA[(wid * 16 + ln) * ASTRIDE + 2 * hi];
        const float2* sW2 = reinterpret_cast<const float2*>(sW);
        #pragma unroll
        for (int s = 0; s < 16; ++s) {
            int kk = s * 4;
            // A frag (16x4 f32): v[0]=A[m][kk+2*hi], v[1]=A[m][kk+2*hi+1]
            float2 a2 = *reinterpret_cast<const float2*>(aRow + kk);
            v2f a; a.x = a2.x; a.y = a2.y;
            #pragma unroll
            for (int n = 0; n < 4; ++n) {
                int nn = n * 16 + ln;
                // B frag rows (kk+2*hi, kk+2*hi+1) at col nn: one aligned float2
                float2 b2 = sW2[(2 * s + hi) * 64 + nn];
                v2f b; b.x = b2.x; b.y = b2.y;
                acc[n] = __builtin_amdgcn_wmma_f32_16x16x4_f32(
                    false, a, false, b, (short)0, acc[n], false, false);
            }
        }
    }

    // store pre-BN result: C/D layout: VGPR r -> M=r (lanes 0-15), M=r+8 (lanes 16-31); N=ln
    const int mBase = blockIdx.x * 128 + wid * 16 + 8 * hi;
    #pragma unroll
    for (int r = 0; r < 8; ++r) {
        int m = mBase + r;
        int b = m / HW;
        int rr = m % HW;
        float* ob = out + b * (COUT * HW) + rr;
        #pragma unroll
        for (int n = 0; n < 4; ++n)
            ob[(n * 16 + ln) * HW] = acc[n][r];
    }
}

// ---------------- Kernel 4: BN stats (one WG per channel) ----------------
__global__ void k_bn_stats(const float* __restrict__ out, const float* __restrict__ gamma,
                           float* __restrict__ stats) {
    __shared__ double ssum[256];
    __shared__ double ssq[256];
    int c = blockIdx.x;
    int tid = threadIdx.x;
    double sum = 0.0, sq = 0.0;
    for (int b = 0; b < NB; ++b) {
        const float* p = out + (b * COUT + c) * HW;
        for (int i = tid; i < HW; i += 256) {
            double v = (double)p[i];
            sum += v;
            sq += v * v;
        }
    }
    ssum[tid] = sum; ssq[tid] = sq;
    __syncthreads();
    for (int s = 128; s > 0; s >>= 1) {
        if (tid < s) { ssum[tid] += ssum[tid + s]; ssq[tid] += ssq[tid + s]; }
        __syncthreads();
    }
    if (tid == 0) {
        double n = (double)MTOT;
        double mean = ssum[0] / n;
        double var = ssq[0] / n - mean * mean;
        stats[c] = (float)mean;
        stats[64 + c] = gamma[c] / sqrtf((float)var + 1e-5f);
    }
}

// ---------------- Kernel 5: BN apply + ReLU in place ----------------
__global__ void k_bn_apply(float* __restrict__ out, const float* __restrict__ stats,
                           const float* __restrict__ beta) {
    int idx = blockIdx.x * 256 + threadIdx.x;
    if (idx >= OUT_ELEMS) return;
    int c = (idx / HW) & 63;
    float v = out[idx];
    float r = (v - stats[c]) * stats[64 + c] + beta[c];
    out[idx] = fmaxf(r, 0.0f);
}

extern "C" void kernel_launch(void* const* d_in, const int* in_sizes, int n_in,
                              void* d_out, int out_size, void* d_ws, size_t ws_size,
                              hipStream_t stream) {
    const float* x        = (const float*)d_in[0];
    const float* d        = (const float*)d_in[1];
    const float* offset_w = (const float*)d_in[2];
    const float* conv_w   = (const float*)d_in[3];
    const float* gamma    = (const float*)d_in[4];
    const float* beta     = (const float*)d_in[5];

    float* out_main = (float*)d_out;                 // [B,64,H,W]
    float* out_off  = out_main + OUT_ELEMS;          // [B,18,H,W]

    float* xh    = (float*)d_ws;                     // 6,553,600 floats (NHWC x)
    float* wt    = xh + OUT_ELEMS;                   // 36,864 floats (interleaved Wt)
    float* stats = wt + 576 * 64;                    // 128 floats (mean | inv)

    k_transpose_nhwc<<<OUT_ELEMS / 256, 256, 0, stream>>>(x, xh);
    k_transpose_w<<<(576 * 64 + 255) / 256, 256, 0, stream>>>(conv_w, wt);
    k_offset_conv<<<OFF_ELEMS / 256, 256, 0, stream>>>(d, offset_w, out_off);
    size_t shbytes = (4096 + 128 * ASTRIDE) * sizeof(float);   // 51,200 B
    k_deform_gemm<<<MTOT / 128, 256, shbytes, stream>>>(xh, wt, out_off, out_main);
    k_bn_stats<<<COUT, 256, 0, stream>>>(out_main, gamma, stats);
    k_bn_apply<<<OUT_ELEMS / 256, 256, 0, stream>>>(out_main, stats, beta);
}